// SO3Convolution_29368986370623
// MI455X (gfx1250) — compile-verified
//
#include <hip/hip_runtime.h>

// ---------------------------------------------------------------------------
// SO3 convolution for MI455X (gfx1250, wave32)
//   Kernel A: Wfull = (radial @ W + b) * cutoff  via V_WMMA_F32_16X16X4_F32
//   Kernel B: per-atom gather (idx_i sorted) of sparse CG contraction,
//             deterministic (no atomics). Entries grouped by idx_out so
//             accumulators live in registers; dir*W pre-fused per edge.
// ---------------------------------------------------------------------------

typedef float v2f __attribute__((ext_vector_type(2)));
typedef float v8f __attribute__((ext_vector_type(8)));

#define S9   9        // (LMAX+1)^2
#define NF   64       // features
#define NCOL 192      // (LMAX+1)*NF
#define NK   20       // radial basis
#define MAXNNZ 768    // >= 9*9*9 is impossible to exceed after CG sparsify

// ---------------------------------------------------------------------------
// Kernel A: GEMM (E x 20) @ (20 x 192) with f32 WMMA, epilogue +b, *cutoff.
// One wave computes one 16x16 tile; K=20 -> 5 steps of K=4.
// A layout (16x4 f32): lane<16: M=lane, holds K=k0,k0+1 ; lane>=16: K=k0+2,k0+3
// B layout (4x16 f32): lane<16: N=lane, holds K=k0,k0+1 ; lane>=16: K=k0+2,k0+3
// C/D (16x16 f32, 8 VGPRs): vgpr r -> M = r + (lane<16 ? 0 : 8), N = lane%16
// ---------------------------------------------------------------------------
__global__ __launch_bounds__(256)
void so3_radial_gemm_wmma(const float* __restrict__ radial,
                          const float* __restrict__ Wmat,
                          const float* __restrict__ bvec,
                          const float* __restrict__ cutoff,
                          float* __restrict__ wfull,
                          int nEdges)
{
    const int NT   = NCOL / 16;                       // 12 N-tiles
    int wave  = blockIdx.x * (blockDim.x >> 5) + (threadIdx.x >> 5);
    int lane  = threadIdx.x & 31;
    int tileM = wave / NT;
    int tileN = wave - tileM * NT;
    int m0    = tileM * 16;
    int n0    = tileN * 16;
    if (m0 >= nEdges) return;

    int half = lane >> 4;          // 0: K pair (k0,k0+1), 1: (k0+2,k0+3)
    int ln   = lane & 15;

    v8f acc = {0.f, 0.f, 0.f, 0.f, 0.f, 0.f, 0.f, 0.f};

#pragma unroll
    for (int kk = 0; kk < 5; ++kk) {
        int k0 = kk * 4 + 2 * half;
        // A: 2 consecutive K values of row m0+ln (rows are 20 floats -> even
        // float index => 8B aligned, safe float2 load)
        v2f a = *(const v2f*)(radial + (size_t)(m0 + ln) * NK + k0);
        // B: column n0+ln, rows k0 and k0+1
        v2f b;
        b.x = Wmat[(size_t)k0 * NCOL + n0 + ln];
        b.y = Wmat[(size_t)(k0 + 1) * NCOL + n0 + ln];
        acc = __builtin_amdgcn_wmma_f32_16x16x4_f32(
            /*neg_a=*/false, a, /*neg_b=*/false, b,
            /*c_mod=*/(short)0, acc, /*reuse_a=*/false, /*reuse_b=*/false);
    }

    float bn = bvec[n0 + ln];
#pragma unroll
    for (int r = 0; r < 8; ++r) {
        int m = m0 + r + 8 * half;
        float v = (acc[r] + bn) * cutoff[m];
        wfull[(size_t)m * NCOL + n0 + ln] = v;
    }
}

// ---------------------------------------------------------------------------
// Kernel B
// ---------------------------------------------------------------------------
__device__ __forceinline__ int lower_bound_i32(const int* __restrict__ a,
                                               int n, int key)
{
    int lo = 0, hi = n;
    while (lo < hi) {
        int mid = (lo + hi) >> 1;
        if (a[mid] < key) lo = mid + 1; else hi = mid;
    }
    return lo;
}

__global__ __launch_bounds__(256)
void so3_cg_gather(const float* __restrict__ x,
                   const float* __restrict__ dir,
                   const float* __restrict__ wfull,
                   const float* __restrict__ radial,
                   const float* __restrict__ Wmat,
                   const float* __restrict__ bvec,
                   const float* __restrict__ cutoff,
                   const int*   __restrict__ idx_i,
                   const int*   __restrict__ idx_j,
                   const float* __restrict__ cg,
                   const int*   __restrict__ in1,
                   const int*   __restrict__ in2,
                   const int*   __restrict__ iout,
                   int nnz, int nEdges, int useWfull,
                   float* __restrict__ y)
{
    // per-thread LDS columns [slot][tid]: tid fastest -> bank-conflict free,
    // and each column is private to its thread (no barriers needed in loop).
    __shared__ float  dw[S9 * 256];     // dir[s] * W_{l(s)} per thread
    __shared__ float  xv[S9 * 256];     // x[idx_j] row per thread
    __shared__ float2 tab[MAXNNZ];      // {cg, in1|in2<<8}, grouped by idx_out
    __shared__ int    off[S9 + 1];
    __shared__ int    cnts[S9];

    int tid  = threadIdx.x;
    int f    = tid & 63;
    int sub  = tid >> 6;               // 0..3 edge subgroups
    int atom = blockIdx.x;
    int nz   = (nnz > MAXNNZ) ? MAXNNZ : nnz;

    // ---- build idx_out-grouped table (deterministic counting sort) ----
    if (tid < S9) {
        int c = 0;
        for (int p = 0; p < nz; ++p) c += (iout[p] == tid);
        cnts[tid] = c;
    }
    __syncthreads();
    if (tid == 0) {
        int r = 0;
        off[0] = 0;
#pragma unroll
        for (int o = 0; o < S9; ++o) { r += cnts[o]; off[o + 1] = r; }
    }
    __syncthreads();
    if (tid < S9) {
        int w = off[tid];
        for (int p = 0; p < nz; ++p) {
            if (iout[p] == tid) {
                float2 t;
                t.x = cg[p];
                t.y = __int_as_float(in1[p] | (in2[p] << 8));
                tab[w++] = t;
            }
        }
    }
    __syncthreads();

    int offr[S9 + 1];
#pragma unroll
    for (int o = 0; o <= S9; ++o) offr[o] = off[o];

    int lo = lower_bound_i32(idx_i, nEdges, atom);
    int hi = lower_bound_i32(idx_i, nEdges, atom + 1);

    float yreg[S9];
#pragma unroll
    for (int s = 0; s < S9; ++s) yreg[s] = 0.f;

    for (int e = lo + sub; e < hi; e += 4) {
        int j = idx_j[e];

        float w0, w1, w2;
        if (useWfull) {
            w0 = wfull[(size_t)e * NCOL + 0 * NF + f];
            w1 = wfull[(size_t)e * NCOL + 1 * NF + f];
            w2 = wfull[(size_t)e * NCOL + 2 * NF + f];
        } else {
            float cut = cutoff[e];
            float a0 = bvec[f], a1 = bvec[NF + f], a2 = bvec[2 * NF + f];
            for (int k = 0; k < NK; ++k) {
                float r = radial[(size_t)e * NK + k];
                a0 += r * Wmat[(size_t)k * NCOL + f];
                a1 += r * Wmat[(size_t)k * NCOL + NF + f];
                a2 += r * Wmat[(size_t)k * NCOL + 2 * NF + f];
            }
            w0 = a0 * cut; w1 = a1 * cut; w2 = a2 * cut;
        }

        // stage fused dir*W and gathered x row into this thread's LDS columns
#pragma unroll
        for (int s = 0; s < S9; ++s) {
            float wl = (s == 0) ? w0 : ((s < 4) ? w1 : w2);   // l(s) map
            dw[s * 256 + tid] = dir[(size_t)e * S9 + s] * wl;
            xv[s * 256 + tid] = x[(size_t)j * (S9 * NF) + s * NF + f];
        }

        // grouped sparse contraction: accumulators stay in registers
#pragma unroll
        for (int o = 0; o < S9; ++o) {
            float acc = 0.f;
            for (int p = offr[o]; p < offr[o + 1]; ++p) {
                float2 t = tab[p];                       // ds_load_b64 broadcast
                int ix = __float_as_int(t.y);
                acc += t.x * dw[(ix & 255) * 256 + tid]
                           * xv[((ix >> 8) & 255) * 256 + tid];
            }
            yreg[o] += acc;
        }
    }

    // park per-thread results in own LDS column (reuse dw), then reduce subs
#pragma unroll
    for (int s = 0; s < S9; ++s) dw[s * 256 + tid] = yreg[s];
    __syncthreads();
    if (sub == 0) {
#pragma unroll
        for (int s = 0; s < S9; ++s) {
            float v = dw[s * 256 + f]       + dw[s * 256 + 64 + f]
                    + dw[s * 256 + 128 + f] + dw[s * 256 + 192 + f];
            y[(size_t)atom * (S9 * NF) + s * NF + f] = v;
        }
    }
}

// ---------------------------------------------------------------------------
extern "C" void kernel_launch(void* const* d_in, const int* in_sizes, int n_in,
                              void* d_out, int out_size, void* d_ws, size_t ws_size,
                              hipStream_t stream)
{
    const float* x       = (const float*)d_in[0];   // (A, 9, 64)
    const float* radial  = (const float*)d_in[1];   // (E, 20)
    const float* dir     = (const float*)d_in[2];   // (E, 9)
    const float* cutoff  = (const float*)d_in[3];   // (E, 1)
    const int*   idx_i   = (const int*)  d_in[4];   // (E,) sorted
    const int*   idx_j   = (const int*)  d_in[5];   // (E,)
    const float* Wmat    = (const float*)d_in[6];   // (20, 192)
    const float* bvec    = (const float*)d_in[7];   // (192,)
    const float* cg      = (const float*)d_in[8];   // (nnz,)
    const int*   in1     = (const int*)  d_in[9];
    const int*   in2     = (const int*)  d_in[10];
    const int*   iout    = (const int*)  d_in[11];
    // d_in[12] (widx) not needed: widx[p] == l(in1[p]), fused at staging time
    float*       y       = (float*)d_out;

    int nEdges = in_sizes[3];                 // cutoff has E elements
    int nnz    = in_sizes[8];
    int nAtoms = in_sizes[0] / (S9 * NF);

    size_t need = (size_t)nEdges * NCOL * sizeof(float);
    int useWfull = (ws_size >= need) ? 1 : 0;
    float* wfull = (float*)d_ws;

    if (useWfull) {
        int mTiles = (nEdges + 15) / 16;
        int waves  = mTiles * (NCOL / 16);
        int blocks = (waves + 7) / 8;         // 8 waves (256 threads) per block
        so3_radial_gemm_wmma<<<blocks, 256, 0, stream>>>(
            radial, Wmat, bvec, cutoff, wfull, nEdges);
    }

    so3_cg_gather<<<nAtoms, 256, 0, stream>>>(
        x, dir, wfull, radial, Wmat, bvec, cutoff,
        idx_i, idx_j, cg, in1, in2, iout,
        nnz, nEdges, useWfull, y);
}